// ROIPoolingLayer_33071248179308
// MI455X (gfx1250) — compile-verified
//
#include <hip/hip_runtime.h>

// ROI max pooling (NHWC), tuned for MI455X / gfx1250 (wave32).
//
// Roofline: feature map (10 MB) is L2-resident (192 MB L2); reads ~100 MB
// served from L2, HBM traffic ~20 MB total -> latency-bound, not BW-bound.
// No multiply-accumulate exists in this op, so WMMA is inapplicable; the
// winning CDNA5 paths are coalesced global_load_b128 (1 KB/txn), scalarized
// ROI math on the SALU, global_prefetch_b8 row lookahead, and non-temporal
// b128 stores so the streamed-once output doesn't evict the hot feature map.
//
// This revision widens the reduction to 2 rows x 2 cols per inner iteration
// (4 independent load->max chains) to hide L2 latency inside the tiny
// (typ. 2-4 x 2-4) bin windows.

typedef float v4f __attribute__((ext_vector_type(4)));

#define FM_H 50
#define FM_W 50
#define FM_C 256
#define PH 7
#define PW 7
#define CVEC (FM_C / 4)   // 64 float4 groups per (y,x) position

static __device__ __forceinline__ v4f vmax4(v4f a, v4f b) {
  v4f r;
  r.x = __builtin_fmaxf(a.x, b.x);
  r.y = __builtin_fmaxf(a.y, b.y);
  r.z = __builtin_fmaxf(a.z, b.z);
  r.w = __builtin_fmaxf(a.w, b.w);
  return r;
}

__global__ __launch_bounds__(64) void roi_maxpool_gfx1250(
    const float* __restrict__ fm,    // (B, H, W, C)
    const float* __restrict__ rois,  // (B, R, 4): y0, x0, y1, x1 in [0,1]
    float* __restrict__ out,         // (B, R, 7, 7, C)
    int R)
{
  const int bid = blockIdx.x;                 // (b*R + r)*49 + ph*7 + pw
  const int bin = bid % (PH * PW);
  const int br  = bid / (PH * PW);
  const int pw  = bin % PW;
  const int ph  = bin / PW;
  const int b   = br / R;

  // ROI -> integer window (f32 multiply + truncation, matches reference).
  const float* roi = rois + (size_t)br * 4;
  const int hs = (int)((float)FM_H * roi[0]);
  const int ws = (int)((float)FM_W * roi[1]);
  const int he = (int)((float)FM_H * roi[2]);
  const int we = (int)((float)FM_W * roi[3]);
  int sh = (he - hs) / PH; if (sh < 1) sh = 1;
  int sw = (we - ws) / PW; if (sw < 1) sw = 1;

  // Bin row/col range; last bin absorbs remainder (== clip((i-s)//st, 0, 6)).
  const int yb0 = hs + ph * sh;
  int yb1 = hs + (ph + 1) * sh; if (ph == PH - 1 || yb1 > he) yb1 = he;
  const int xb0 = ws + pw * sw;
  int xb1 = ws + (pw + 1) * sw; if (pw == PW - 1 || xb1 > we) xb1 = we;

  const int lane = (int)threadIdx.x;          // 0..63, 4 channels per lane
  const v4f* fm4 = (const v4f*)fm;
  const size_t rowpitch = (size_t)FM_W * CVEC;
  const v4f* img = fm4 + (size_t)b * FM_H * rowpitch + (size_t)lane;

  const float ninf = -__builtin_inff();
  v4f m0 = { ninf, ninf, ninf, ninf };
  v4f m1 = m0, m2 = m0, m3 = m0;              // 4 accumulators -> 16 independent
                                              // v_max_num_f32 chains per lane
  int y = yb0;
  // Two rows per iteration: 4 independent b128 loads in flight.
  for (; y + 1 < yb1; y += 2) {
    const v4f* rowA = img + (size_t)y * rowpitch;
    const v4f* rowB = rowA + rowpitch;
    if (y + 2 < yb1) {
      // gfx1250: global_prefetch_b8 — pull the row pair after this one
      __builtin_prefetch((const void*)(rowB + rowpitch + (size_t)xb0 * CVEC),
                         0, 0);
    }
    int x = xb0;
    for (; x + 1 < xb1; x += 2) {
      v4f a0 = rowA[(size_t)x * CVEC];
      v4f a1 = rowA[(size_t)(x + 1) * CVEC];
      v4f b0 = rowB[(size_t)x * CVEC];
      v4f b1 = rowB[(size_t)(x + 1) * CVEC];
      m0 = vmax4(m0, a0);
      m1 = vmax4(m1, a1);
      m2 = vmax4(m2, b0);
      m3 = vmax4(m3, b1);
    }
    if (x < xb1) {
      m0 = vmax4(m0, rowA[(size_t)x * CVEC]);
      m2 = vmax4(m2, rowB[(size_t)x * CVEC]);
    }
  }
  // Remainder row (odd bin height).
  if (y < yb1) {
    const v4f* rowA = img + (size_t)y * rowpitch;
    int x = xb0;
    for (; x + 1 < xb1; x += 2) {
      v4f a0 = rowA[(size_t)x * CVEC];
      v4f a1 = rowA[(size_t)(x + 1) * CVEC];
      m0 = vmax4(m0, a0);
      m1 = vmax4(m1, a1);
    }
    if (x < xb1) m0 = vmax4(m0, rowA[(size_t)x * CVEC]);
  }
  m0 = vmax4(vmax4(m0, m1), vmax4(m2, m3));

  // Streamed-once output: non-temporal b128 store (keep L2 for feature map).
  v4f* out4 = (v4f*)out;
  __builtin_nontemporal_store(m0, out4 + (size_t)bid * CVEC + (size_t)lane);
}

extern "C" void kernel_launch(void* const* d_in, const int* in_sizes, int n_in,
                              void* d_out, int out_size, void* d_ws, size_t ws_size,
                              hipStream_t stream) {
  const float* fm   = (const float*)d_in[0];   // feature_map (B,H,W,C) f32
  const float* rois = (const float*)d_in[1];   // rois (B,R,4) f32
  float* out = (float*)d_out;                  // (B,R,7,7,C) f32

  const int nroi_total = in_sizes[1] / 4;                  // B*R (=200)
  const int B = in_sizes[0] / (FM_H * FM_W * FM_C);        // (=2)
  const int R = nroi_total / (B > 0 ? B : 1);              // (=100)
  const int grid = nroi_total * PH * PW;                   // (=9800 blocks)

  roi_maxpool_gfx1250<<<grid, 64, 0, stream>>>(fm, rois, out, R);
}